// PostProcessor_2207613190144
// MI455X (gfx1250) — compile-verified
//
#include <hip/hip_runtime.h>
#include <stdint.h>
#include <math.h>

// ---------------- problem constants (from reference) ----------------
#define BATCH   32
#define CCH     3
#define HH      96
#define WW      320
#define HW      (HH * WW)          // 30720
#define CHW     (CCH * HW)         // 92160
#define RCH     50
#define KDET    100
#define NDET    (BATCH * KDET)     // 3200
#define NTHR    1024
#define STRIP   16
#define NSTRIPS (HH / STRIP)       // 6
#define TROWS   (STRIP + 2)        // 18 (1-row halo each side)
#define TELEMS  (TROWS * WW)       // 5760
#define TV4     (TELEMS / 4)       // 1440 float4 tiles
#define WV4     (WW / 4)           // 80 float4 per row
#define KPS     ((STRIP * WW) / NTHR) // 5 pixels per thread per strip
#define SLOTS   (CCH * NSTRIPS * KPS) // 90 elements owned per thread
#define PI_F    3.14159265358979323846f
#define NEG_INF (-3.402823466e38f)

// ---------------- CDNA5 async global->LDS copy path ----------------
// GVS form: saddr = uniform 64-bit base (SGPR pair), vaddr = per-lane
// 32-bit byte offset, vdst = LDS byte address. Tracked by ASYNCcnt.
__device__ __forceinline__ void async_load_b128(unsigned lds_byte_addr,
                                                const void* base,
                                                unsigned byte_off) {
  asm volatile("global_load_async_to_lds_b128 %0, %1, %2"
               :: "v"(lds_byte_addr), "v"(byte_off),
                  "s"((unsigned long long)(uintptr_t)base)
               : "memory");
}
__device__ __forceinline__ void wait_async0() {
  asm volatile("s_wait_asynccnt 0x0" ::: "memory");
}

__device__ __forceinline__ float wrapf(float a) {
  float r = fmodf(a + PI_F, 2.0f * PI_F);
  if (r < 0.0f) r += 2.0f * PI_F;   // Python-style modulo
  return r - PI_F;
}

// =====================================================================
// Kernel 1: 3x3 NMS (LDS-staged via async b128 copies) + per-batch
// top-100. One 1024-thread block per batch element.
// =====================================================================
__global__ __launch_bounds__(NTHR) void nms_topk_kernel(
    const float* __restrict__ heat,     // [B, C, H, W]
    float* __restrict__ wsv,            // [B, CHW] nms scores, slot-major/tid-minor
    float* __restrict__ tscore,         // [B*K]
    int*   __restrict__ tidx)           // [B*K] flat index in [0, CHW)
{
  __shared__ __align__(16) float tile[TELEMS];
  __shared__ float cval[NTHR];
  __shared__ int   cidx[NTHR];
  __shared__ float rval[NTHR];
  __shared__ int   ridx[NTHR];

  const int b   = blockIdx.x;
  const int tid = threadIdx.x;
  const unsigned tile_base = (unsigned)(uintptr_t)(&tile[0]); // low 32 bits = LDS offset
  const float* hb = heat + (size_t)b * CHW;

  float bestV = -1.0f;
  int   bestI = 0;

  for (int c = 0; c < CCH; ++c) {
    for (int s = 0; s < NSTRIPS; ++s) {
      const int r0 = s * STRIP;
      // ---- stage strip (with halo rows) into LDS, 16B per async op
      for (int slot = tid; slot < TV4; slot += NTHR) {
        int trow = slot / WV4;           // 0..17
        int col4 = slot - trow * WV4;    // 0..79
        int grow = r0 - 1 + trow;
        if (grow >= 0 && grow < HH) {
          unsigned goff = (unsigned)((((c * HH + grow) * WW) + col4 * 4) * 4);
          async_load_b128(tile_base + (unsigned)slot * 16u, hb, goff);
        } else {
          float4 n4; n4.x = NEG_INF; n4.y = NEG_INF; n4.z = NEG_INF; n4.w = NEG_INF;
          ((float4*)tile)[slot] = n4;    // SAME padding == -inf under max
        }
      }
      wait_async0();
      __syncthreads();

      // ---- 3x3 max-pool NMS from LDS; write scores coalesced to ws
      #pragma unroll
      for (int k = 0; k < KPS; ++k) {
        int p    = k * NTHR + tid;       // 0..5119
        int orow = p / WW;               // 0..15
        int col  = p - orow * WW;
        int tr   = orow + 1;
        float center = tile[tr * WW + col];
        float m = center;
        int c0 = (col > 0)      ? col - 1 : col;
        int c1 = (col < WW - 1) ? col + 1 : col;
        #pragma unroll
        for (int dr = -1; dr <= 1; ++dr) {
          const float* rp = &tile[(tr + dr) * WW];
          for (int cc = c0; cc <= c1; ++cc) m = fmaxf(m, rp[cc]);
        }
        float val = (center == m) ? center : 0.0f;
        int gidx  = c * HW + (r0 + orow) * WW + col;
        int slot  = (c * NSTRIPS + s) * KPS + k;
        wsv[(size_t)b * CHW + (size_t)slot * NTHR + tid] = val;
        if (val > bestV || (val == bestV && gidx < bestI)) { bestV = val; bestI = gidx; }
      }
      __syncthreads();   // protect tile before next strip's fill
    }
  }

  // ---- top-K by iterative block argmax over per-thread chunk maxima
  cval[tid] = bestV; cidx[tid] = bestI;
  __syncthreads();

  for (int k = 0; k < KDET; ++k) {
    rval[tid] = cval[tid]; ridx[tid] = cidx[tid];
    __syncthreads();
    for (int off = NTHR >> 1; off > 0; off >>= 1) {
      if (tid < off) {
        float ov = rval[tid + off]; int oi = ridx[tid + off];
        float mv = rval[tid];       int mi = ridx[tid];
        if (ov > mv || (ov == mv && oi < mi)) { rval[tid] = ov; ridx[tid] = oi; }
      }
      __syncthreads();
    }
    int wi = ridx[0];
    if (tid == 0) { tscore[b * KDET + k] = rval[0]; tidx[b * KDET + k] = wi; }

    // owning thread invalidates the extracted element and rescans its chunk
    int wc   = wi / HW;
    int rem  = wi - wc * HW;
    int wrow = rem / WW;
    int wcol = rem - wrow * WW;
    int wss  = wrow / STRIP;
    int wp   = (wrow - wss * STRIP) * WW + wcol;
    int wk   = wp / NTHR;
    int owner = wp - wk * NTHR;
    if (tid == owner) {
      int wslot = (wc * NSTRIPS + wss) * KPS + wk;
      size_t basea = (size_t)b * CHW + (size_t)tid;
      wsv[basea + (size_t)wslot * NTHR] = -1.0f;  // below any score (>=0)
      float nb = -1.0f; int ni = 0;
      for (int slot = 0; slot < SLOTS; ++slot) {
        float v = wsv[basea + (size_t)slot * NTHR];
        int sc   = slot / (NSTRIPS * KPS);
        int srem = slot - sc * (NSTRIPS * KPS);
        int ss   = srem / KPS;
        int sk   = srem - ss * KPS;
        int pp   = sk * NTHR + tid;
        int orow2 = pp / WW;
        int col2  = pp - orow2 * WW;
        int gidx2 = sc * HW + (ss * STRIP + orow2) * WW + col2;
        if (v > nb || (v == nb && gidx2 < ni)) { nb = v; ni = gidx2; }
      }
      cval[tid] = nb; cidx[tid] = ni;
    }
    __syncthreads();
  }
}

// =====================================================================
// Kernel 2: gather 50 regression channels at top-K indices and decode.
// =====================================================================
__global__ void decode_kernel(
    const float* __restrict__ reg,        // [B, R, HW]
    const float* __restrict__ dims_mean,  // [3,3]
    const float* __restrict__ calib,      // [B,3,4]
    const float* __restrict__ pad_size,   // [B,2]
    const float* __restrict__ img_size,   // [B,2]
    const float* __restrict__ tscore,     // [N]
    const int*   __restrict__ tidx,       // [N]
    float* __restrict__ result,           // [N,14]
    float* __restrict__ pred7)            // [N,7]
{
  int t = blockIdx.x * blockDim.x + threadIdx.x;
  if (t >= NDET) return;
  const int b = t / KDET;

  float score = tscore[t];
  int   idx   = tidx[t];
  int   cls   = idx / HW;
  int   ind   = idx - cls * HW;
  int   iy    = ind / WW;
  int   ix    = ind - iy * WW;
  float ys = (float)iy, xs = (float)ix;

  float p[RCH];
  const float* rb = reg + (size_t)b * RCH * HW + (size_t)ind;
  #pragma unroll
  for (int r = 0; r < RCH; ++r) p[r] = rb[(size_t)r * HW];

  const float pad0 = pad_size[b * 2 + 0], pad1 = pad_size[b * 2 + 1];
  const float imw  = img_size[b * 2 + 0], imh  = img_size[b * 2 + 1];
  const float f  = calib[b * 12 + 0];
  const float cx = calib[b * 12 + 2];
  const float cy = calib[b * 12 + 6];

  // 2D box
  float r0 = fmaxf(p[0], 0.f), r1 = fmaxf(p[1], 0.f);
  float r2 = fmaxf(p[2], 0.f), r3 = fmaxf(p[3], 0.f);
  float x1 = (xs - r0) * 4.f - pad0;
  float y1 = (ys - r1) * 4.f - pad1;
  float x2 = (xs + r2) * 4.f - pad0;
  float y2 = (ys + r3) * 4.f - pad1;
  x1 = fminf(fmaxf(x1, 0.f), imw - 1.f);
  y1 = fminf(fmaxf(y1, 0.f), imh - 1.f);
  x2 = fminf(fmaxf(x2, 0.f), imw - 1.f);
  y2 = fminf(fmaxf(y2, 0.f), imh - 1.f);

  // dims
  float d0 = dims_mean[cls * 3 + 0] * expf(p[29]);
  float d1 = dims_mean[cls * 3 + 1] * expf(p[30]);
  float d2 = dims_mean[cls * 3 + 2] * expf(p[31]);

  // depth ensemble
  float sig = 1.f / (1.f + expf(-p[48]));
  float dd  = fminf(fmaxf(1.f / sig - 1.f, 0.1f), 100.f);
  float du  = expf(p[49]);
  float hch = p[23] - p[25];                          // kp[8].y - kp[9].y
  float h02 = (p[7]  + p[11] - (p[15] + p[19])) * 0.5f;
  float h13 = (p[9]  + p[13] - (p[17] + p[21])) * 0.5f;
  float hts[3] = { fmaxf(hch * 4.f, 0.1f), fmaxf(h02 * 4.f, 0.1f), fmaxf(h13 * 4.f, 0.1f) };
  float kd[3], ku[3];
  #pragma unroll
  for (int i = 0; i < 3; ++i) {
    kd[i] = fminf(fmaxf(f * d1 / hts[i], 0.1f), 100.f);
    ku[i] = expf(p[26 + i]);
  }
  float w0 = 1.f / du, w1 = 1.f / ku[0], w2 = 1.f / ku[1], w3 = 1.f / ku[2];
  float wsum = w0 + w1 + w2 + w3;
  float depth = (dd * w0 + kd[0] * w1 + kd[1] * w2 + kd[2] * w3) / wsum;

  // orientation
  float bd = p[33] - p[32]; int bin = 0;
  #pragma unroll
  for (int i = 1; i < 4; ++i) {
    float di = p[32 + 2 * i + 1] - p[32 + 2 * i];
    if (di > bd) { bd = di; bin = i; }
  }
  float so0 = p[40 + 2 * bin], so1 = p[41 + 2 * bin];
  float nrm = sqrtf(so0 * so0 + so1 * so1) + 1e-8f;
  so0 /= nrm; so1 /= nrm;
  const float binc[4] = { 0.f, PI_F * 0.5f, PI_F, -PI_F * 0.5f };
  float alpha = wrapf(binc[bin] + atan2f(so0, so1));

  // 3D location
  float uu = (xs + p[4]) * 4.f - pad0;
  float vv = (ys + p[5]) * 4.f - pad1;
  float x3d = (uu - cx) * depth / f;
  float y3d = (vv - cy) * depth / f + d1 * 0.5f;
  float roty = wrapf(alpha + atan2f(x3d, depth));

  float* out = result + (size_t)t * 14;
  out[0]  = (float)cls; out[1]  = alpha;
  out[2]  = x1;  out[3]  = y1;  out[4]  = x2;  out[5]  = y2;
  out[6]  = d0;  out[7]  = d1;  out[8]  = d2;
  out[9]  = x3d; out[10] = y3d; out[11] = depth;
  out[12] = roty; out[13] = score;

  float* pr = pred7 + (size_t)t * 7;
  pr[0] = x3d; pr[1] = y3d; pr[2] = depth;
  pr[3] = d0;  pr[4] = d1;  pr[5] = d2;  pr[6] = roty;
}

// =====================================================================
// Kernel 3: rotated 3D IoU, exact replica of the reference algorithm.
// =====================================================================
__device__ float rect_overlap(float ax, float ay, float aw, float ah, float ar,
                              float bx, float by, float bw, float bh, float br)
{
  const float lx[4] = { 0.5f, 0.5f, -0.5f, -0.5f };
  const float ly[4] = { 0.5f, -0.5f, -0.5f, 0.5f };
  float cax[4], cay[4], cbx[4], cby[4];
  float c_a = cosf(ar), s_a = sinf(ar);
  float c_b = cosf(br), s_b = sinf(br);
  #pragma unroll
  for (int i = 0; i < 4; ++i) {
    float px = lx[i] * aw, py = ly[i] * ah;
    cax[i] = c_a * px - s_a * py + ax;
    cay[i] = s_a * px + c_a * py + ay;
    px = lx[i] * bw; py = ly[i] * bh;
    cbx[i] = c_b * px - s_b * py + bx;
    cby[i] = s_b * px + c_b * py + by;
  }
  const float eps = 1e-5f;
  float ptx[24], pty[24]; bool msk[24];
  #pragma unroll
  for (int i = 0; i < 4; ++i) {
    ptx[i] = cax[i]; pty[i] = cay[i];
    float dx = cax[i] - bx, dy = cay[i] - by;
    float l0 = dx * c_b + dy * s_b, l1 = -dx * s_b + dy * c_b;
    msk[i] = (fabsf(l0) <= bw * 0.5f + eps) && (fabsf(l1) <= bh * 0.5f + eps);
  }
  #pragma unroll
  for (int i = 0; i < 4; ++i) {
    ptx[4 + i] = cbx[i]; pty[4 + i] = cby[i];
    float dx = cbx[i] - ax, dy = cby[i] - ay;
    float l0 = dx * c_a + dy * s_a, l1 = -dx * s_a + dy * c_a;
    msk[4 + i] = (fabsf(l0) <= aw * 0.5f + eps) && (fabsf(l1) <= ah * 0.5f + eps);
  }
  for (int i = 0; i < 4; ++i) {
    float pax = cax[i], pay = cay[i];
    float d1x = cax[(i + 1) & 3] - pax, d1y = cay[(i + 1) & 3] - pay;
    for (int j = 0; j < 4; ++j) {
      float pbx = cbx[j], pby = cby[j];
      float d2x = cbx[(j + 1) & 3] - pbx, d2y = cby[(j + 1) & 3] - pby;
      float rpx = pbx - pax, rpy = pby - pay;
      float den = d1x * d2y - d1y * d2x;
      float den_s = (fabsf(den) < 1e-9f) ? 1e-9f : den;
      float tt = (rpx * d2y - rpy * d2x) / den_s;
      float uu = (rpx * d1y - rpy * d1x) / den_s;
      bool valid = (fabsf(den) > 1e-9f) &&
                   tt >= 0.f && tt <= 1.f && uu >= 0.f && uu <= 1.f;
      int o = 8 + i * 4 + j;
      ptx[o] = pax + tt * d1x; pty[o] = pay + tt * d1y;
      msk[o] = valid;
    }
  }
  float cnt = 0.f, sx = 0.f, sy = 0.f;
  for (int i = 0; i < 24; ++i) if (msk[i]) { cnt += 1.f; sx += ptx[i]; sy += pty[i]; }
  float dv = fmaxf(cnt, 1.f);
  float cxm = sx / dv, cym = sy / dv;
  float relx[24], rely[24], ang[24];
  for (int i = 0; i < 24; ++i) {
    if (msk[i]) {
      relx[i] = ptx[i] - cxm; rely[i] = pty[i] - cym;
      ang[i]  = atan2f(rely[i], relx[i]);
    } else { relx[i] = 0.f; rely[i] = 0.f; ang[i] = 1e9f; }
  }
  int ord[24];
  for (int i = 0; i < 24; ++i) ord[i] = i;
  for (int i = 1; i < 24; ++i) {           // stable insertion sort
    int oi = ord[i]; float ai = ang[oi];
    int j = i - 1;
    while (j >= 0 && ang[ord[j]] > ai) { ord[j + 1] = ord[j]; --j; }
    ord[j + 1] = oi;
  }
  float area = 0.f;
  for (int i = 0; i < 24; ++i) {
    int a = ord[i], c = ord[(i + 1) % 24];
    area += relx[a] * rely[c] - rely[a] * relx[c];
  }
  area = 0.5f * fabsf(area);
  return (cnt >= 3.f) ? area : 0.f;
}

__global__ void iou3d_kernel(const float* __restrict__ pred7,
                             const float* __restrict__ tgt,
                             float* __restrict__ iou_out)
{
  int t = blockIdx.x * blockDim.x + threadIdx.x;
  if (t >= NDET) return;
  const float* pr = pred7 + (size_t)t * 7;
  const float* tb = tgt   + (size_t)t * 7;
  float minA = -(pr[1] + pr[4] * 0.5f), maxA = -(pr[1] - pr[4] * 0.5f);
  float minB = -(tb[1] + tb[4] * 0.5f), maxB = -(tb[1] - tb[4] * 0.5f);
  float hov = fmaxf(fminf(maxA, maxB) - fmaxf(minA, minB), 0.f);
  float overlap = rect_overlap(pr[0], pr[2], pr[3], pr[5], pr[6],
                               tb[0], tb[2], tb[3], tb[5], tb[6]);
  float ov3 = overlap * hov;
  float areaA = pr[3] * pr[5];
  float areaB = tb[3] * tb[5];
  float h0 = pred7[4];  // reference quirk: max_h_a[0]-min_h_a[0] == pred[0,4]
  float uni = areaA * h0 + areaB * (maxB - minB) - ov3;
  iou_out[t] = ov3 / fmaxf(uni, 1e-8f);
}

// =====================================================================
extern "C" void kernel_launch(void* const* d_in, const int* in_sizes, int n_in,
                              void* d_out, int out_size, void* d_ws, size_t ws_size,
                              hipStream_t stream)
{
  const float* heat      = (const float*)d_in[0];
  const float* reg       = (const float*)d_in[1];
  const float* dims_mean = (const float*)d_in[2];
  const float* calib     = (const float*)d_in[3];
  const float* pad_size  = (const float*)d_in[4];
  const float* img_size  = (const float*)d_in[5];
  const float* tgt       = (const float*)d_in[6];
  (void)in_sizes; (void)n_in; (void)out_size; (void)ws_size; // K fixed at 100

  char*  ws     = (char*)d_ws;
  float* wsv    = (float*)ws;                                   // B*CHW floats
  float* tscore = (float*)(ws + (size_t)BATCH * CHW * 4);       // N floats
  int*   tidx   = (int*)((char*)tscore + (size_t)NDET * 4);     // N ints
  float* pred7  = (float*)((char*)tidx + (size_t)NDET * 4);     // N*7 floats

  float* result = (float*)d_out;                 // [N,14]
  float* iou    = result + (size_t)NDET * 14;    // [N]

  nms_topk_kernel<<<BATCH, NTHR, 0, stream>>>(heat, wsv, tscore, tidx);
  decode_kernel<<<(NDET + 255) / 256, 256, 0, stream>>>(
      reg, dims_mean, calib, pad_size, img_size, tscore, tidx, result, pred7);
  iou3d_kernel<<<(NDET + 255) / 256, 256, 0, stream>>>(pred7, tgt, iou);
}